// AtomicConv_51049981280657
// MI455X (gfx1250) — compile-verified
//
#include <hip/hip_runtime.h>
#include <stdint.h>

#define NKER 8
#define NT 6
#define PI_F 3.14159265358979323846f
#define THREADS 256
#define EPT 4                       // edges per thread
#define TILE (THREADS * EPT)        // edges per block

#if __has_builtin(__builtin_amdgcn_global_load_async_to_lds_b128) && \
    __has_builtin(__builtin_amdgcn_s_wait_asynccnt)
#define HAVE_GFX1250_ASYNC 1
#else
#define HAVE_GFX1250_ASYNC 0
#endif

// Hardware f32 atomic add (global_atomic_add_f32), no CAS loop.
__device__ __forceinline__ void atomic_add_f32(float* p, float v) {
    unsafeAtomicAdd(p, v);
}

#if HAVE_GFX1250_ASYNC
typedef int v4i __attribute__((ext_vector_type(4)));
typedef __attribute__((address_space(1))) v4i gmem_v4i;  // prints as "__device__"
typedef __attribute__((address_space(3))) v4i lds_v4i;   // prints as "__shared__"
// Async global->LDS B128 copy, tracked by ASYNCcnt.
__device__ __forceinline__ void async_copy_b128(const void* g, void* l) {
    __builtin_amdgcn_global_load_async_to_lds_b128(
        (gmem_v4i*)g, (lds_v4i*)l, /*offset=*/0, /*cpol=*/0);
}
#endif

// ---------------------------------------------------------------------------
// Pass 1: per-node feature -> one-hot slot code (1 byte), plus a 32-float
// param block: [c[8] | pi/c[8] | r[8] | -g[8]] so the edge kernel needs no
// per-thread divisions and scalar-loads all uniforms.
// ---------------------------------------------------------------------------
__global__ void atomicconv_prep(const float* __restrict__ feat,
                                const float* __restrict__ cutoffs,
                                const float* __restrict__ means,
                                const float* __restrict__ scaling,
                                const float* __restrict__ f2u,
                                float* __restrict__ P,
                                unsigned char* __restrict__ code,
                                int N) {
    int i = blockIdx.x * blockDim.x + threadIdx.x;
    if (i < NKER) {
        float c = cutoffs[i];
        P[i]          = c;
        P[NKER + i]   = PI_F / c;
        P[2*NKER + i] = means[i];
        P[3*NKER + i] = -scaling[i];
    }
    if (i < N) {
        float f = feat[i];
        int t = 0xFF;
#pragma unroll
        for (int j = 0; j < NT; ++j) {
            if (f == f2u[j]) t = j;   // f2u[j]: uniform -> s_load
        }
        code[i] = (unsigned char)t;
    }
}

// ---------------------------------------------------------------------------
// Per-edge envelope + scatter. Shared by main and fallback kernels.
// ---------------------------------------------------------------------------
__device__ __forceinline__ void process_edge(float d, int t, int ddst,
                                             const float* __restrict__ c,
                                             const float* __restrict__ ic,
                                             const float* __restrict__ r,
                                             const float* __restrict__ ng,
                                             float* __restrict__ out) {
    float* row = out + (size_t)ddst * (NT * NKER) + t * NKER;
#pragma unroll
    for (int k = 0; k < NKER; ++k) {
        float dr  = d - r[k];
        float rbf = __expf(ng[k] * dr * dr);          // v_exp_f32
        float cw  = 0.0f;
        if (d <= c[k]) cw = 0.5f * (__cosf(d * ic[k]) + 1.0f);  // v_cos_f32
        atomic_add_f32(row + k, rbf * cw);            // EXEC-masked per lane
    }
}

// ---------------------------------------------------------------------------
// Pass 2: 4 edges per thread. Full tiles are staged global->LDS with the
// CDNA5 async copy path (ASYNCcnt) and consumed via ds_load_b128; each
// thread reads back only its own slots, so only the ASYNCcnt drain is
// needed (no barrier). Atomics fire only for the ~1/3 matching lanes.
// ---------------------------------------------------------------------------
__global__ void __launch_bounds__(THREADS)
atomicconv_edges(const float* __restrict__ dist,
                 const int* __restrict__ src,
                 const int* __restrict__ dst,
                 const unsigned char* __restrict__ code,
                 const float* __restrict__ P,
                 float* __restrict__ out,
                 int E) {
#if HAVE_GFX1250_ASYNC
    __shared__ float sD[TILE];
    __shared__ int   sS[TILE];
    __shared__ int   sT[TILE];
#endif
    float c[NKER], ic[NKER], r[NKER], ng[NKER];
#pragma unroll
    for (int k = 0; k < NKER; ++k) {    // uniform -> scalarized loads
        c[k]  = P[k];
        ic[k] = P[NKER + k];
        r[k]  = P[2*NKER + k];
        ng[k] = P[3*NKER + k];
    }

    const int tid = threadIdx.x;
    const int e0  = (blockIdx.x * THREADS + tid) * EPT;
    if (e0 >= E) return;

    // Speculative prefetch 8192 edges (8 tiles) ahead (gfx1250 global_prefetch_b8;
    // OOB translations are silently dropped).
    __builtin_prefetch(dist + e0 + 8192, 0, 0);
    __builtin_prefetch(src  + e0 + 8192, 0, 0);
    __builtin_prefetch(dst  + e0 + 8192, 0, 0);

    if (e0 + EPT <= E) {
#if HAVE_GFX1250_ASYNC
        async_copy_b128(dist + e0, &sD[tid * EPT]);
        async_copy_b128(src  + e0, &sS[tid * EPT]);
        async_copy_b128(dst  + e0, &sT[tid * EPT]);
        __builtin_amdgcn_s_wait_asynccnt(0);          // drain ASYNCcnt
        float4 d4 = *reinterpret_cast<const float4*>(&sD[tid * EPT]);
        int4   s4 = *reinterpret_cast<const int4*>(&sS[tid * EPT]);
        int4   t4 = *reinterpret_cast<const int4*>(&sT[tid * EPT]);
#else
        float4 d4 = *reinterpret_cast<const float4*>(dist + e0);
        int4   s4 = *reinterpret_cast<const int4*>(src + e0);
        int4   t4 = *reinterpret_cast<const int4*>(dst + e0);
#endif
        float d_[EPT] = {d4.x, d4.y, d4.z, d4.w};
        int   s_[EPT] = {s4.x, s4.y, s4.z, s4.w};
        int   n_[EPT] = {t4.x, t4.y, t4.z, t4.w};
#pragma unroll
        for (int i = 0; i < EPT; ++i) {
            int t = code[s_[i]];
            if (t != 0xFF) process_edge(d_[i], t, n_[i], c, ic, r, ng, out);
        }
    } else {
        for (int e = e0; e < E; ++e) {
            int t = code[src[e]];
            if (t != 0xFF) process_edge(dist[e], t, dst[e], c, ic, r, ng, out);
        }
    }
}

// ---------------------------------------------------------------------------
// Fallback (only if workspace is too small for the code table): single pass,
// inline feature match, per-thread param derivation.
// ---------------------------------------------------------------------------
__global__ void __launch_bounds__(THREADS)
atomicconv_edges_fb(const float* __restrict__ dist,
                    const int* __restrict__ src,
                    const int* __restrict__ dst,
                    const float* __restrict__ feat,
                    const float* __restrict__ cutoffs,
                    const float* __restrict__ means,
                    const float* __restrict__ scaling,
                    const float* __restrict__ f2u,
                    float* __restrict__ out,
                    int E) {
    float c[NKER], ic[NKER], r[NKER], ng[NKER];
#pragma unroll
    for (int k = 0; k < NKER; ++k) {
        c[k]  = cutoffs[k];
        ic[k] = PI_F / c[k];
        r[k]  = means[k];
        ng[k] = -scaling[k];
    }
    int e = blockIdx.x * blockDim.x + threadIdx.x;
    if (e >= E) return;
    float f = feat[src[e]];
    int t = -1;
#pragma unroll
    for (int j = 0; j < NT; ++j) {
        if (f == f2u[j]) t = j;
    }
    if (t < 0) return;
    process_edge(dist[e], t, dst[e], c, ic, r, ng, out);
}

extern "C" void kernel_launch(void* const* d_in, const int* in_sizes, int n_in,
                              void* d_out, int out_size, void* d_ws, size_t ws_size,
                              hipStream_t stream) {
    const float* feat = (const float*)d_in[0];
    const float* dist = (const float*)d_in[1];
    const int*   src  = (const int*)d_in[2];
    const int*   dst  = (const int*)d_in[3];
    const float* cut  = (const float*)d_in[4];
    const float* mean = (const float*)d_in[5];
    const float* scal = (const float*)d_in[6];
    const float* f2u  = (const float*)d_in[7];
    const int N = in_sizes[0];
    const int E = in_sizes[1];
    float* out = (float*)d_out;

    // Zero accumulator every call (graph-capturable async memset).
    (void)hipMemsetAsync(out, 0, (size_t)out_size * sizeof(float), stream);

    const size_t need = 128 + (size_t)N;
    if (ws_size >= need) {
        float* P = (float*)d_ws;
        unsigned char* code = (unsigned char*)d_ws + 128;
        const int pblocks = (N + THREADS - 1) / THREADS;
        atomicconv_prep<<<pblocks, THREADS, 0, stream>>>(feat, cut, mean, scal,
                                                         f2u, P, code, N);
        const int eblocks = (E + TILE - 1) / TILE;
        atomicconv_edges<<<eblocks, THREADS, 0, stream>>>(dist, src, dst, code,
                                                          P, out, E);
    } else {
        const int eblocks = (E + THREADS - 1) / THREADS;
        atomicconv_edges_fb<<<eblocks, THREADS, 0, stream>>>(
            dist, src, dst, feat, cut, mean, scal, f2u, out, E);
    }
}